// Recons_tensor_11321533792292
// MI455X (gfx1250) — compile-verified
//
#include <hip/hip_runtime.h>

// Inverse-unfold (patch reconstruction) + crop, specialized to the reference:
// B=8, C=3, h_patches=w_patches=64, P=16, h_pad=w_pad=8.
//
// in : [B, C, N=4096, 16, 16] f32  (each patch row = 16 contiguous floats)
// out: [B, C, 1016, 1016]     f32  (row-major contiguous)
//
// Pure bandwidth kernel: ~99 MB read + ~99 MB write -> ~8.5 us floor at
// 23.3 TB/s. Zero FLOPs => no WMMA; zero reuse => no LDS/TDM staging.
//
// Mapping: lane = output float4 column (254 active of 256), each thread copies
// 8 consecutive output rows (8 | 16 so all 8 rows live in one patch row:
// hp constant, src advances a fixed 64 B per row, dst 4064 B per row).
// 8 independent b128 NT loads issued back-to-back, then 8 b128 NT stores.

typedef float float4v __attribute__((ext_vector_type(4)));

__global__ __launch_bounds__(256) void recons_tensor_kernel(
    const float* __restrict__ in, float* __restrict__ out)
{
    constexpr int P    = 16;
    constexpr int HP   = 64;          // h_patches
    constexpr int WP   = 64;          // w_patches
    constexpr int HOUT = HP * P - 8;  // 1016
    constexpr int WOUT = WP * P - 8;  // 1016
    constexpr int W4   = WOUT / 4;    // 254 float4s per output row

    const int x4 = threadIdx.x;       // output float4 column
    if (x4 >= W4) return;

    const int ybase = blockIdx.y * 8; // 8 output rows per thread, one patch row
    const int bc    = blockIdx.z;     // fused (b*C + c), 0..23

    const int x   = x4 << 2;          // output column (floats)
    const int wp  = x >> 4;           // patch column
    const int xin = x & 15;           // column inside patch
    const int hp  = ybase >> 4;       // patch row (constant across the 8 rows)
    const int yin = ybase & 15;       // starting row inside patch (0 or 8)

    // src float offset: ((bc*N + hp*WP + wp) * 256) + yin*16 + xin
    const long long srcOff =
        ((long long)(bc * (HP * WP) + hp * WP + wp) << 8) + (yin << 4) + xin;
    // dst float offset: (bc*HOUT + ybase)*WOUT + x
    const long long dstOff = ((long long)(bc * HOUT + ybase)) * WOUT + x;

    const float4v* __restrict__ src = (const float4v*)(in + srcOff);
    float4v* __restrict__ dst       = (float4v*)(out + dstOff);

    // Per row: src += 4 float4s (64 B), dst += 254 float4s (4064 B).
    float4v r0 = __builtin_nontemporal_load(src + 0 * 4);
    float4v r1 = __builtin_nontemporal_load(src + 1 * 4);
    float4v r2 = __builtin_nontemporal_load(src + 2 * 4);
    float4v r3 = __builtin_nontemporal_load(src + 3 * 4);
    float4v r4 = __builtin_nontemporal_load(src + 4 * 4);
    float4v r5 = __builtin_nontemporal_load(src + 5 * 4);
    float4v r6 = __builtin_nontemporal_load(src + 6 * 4);
    float4v r7 = __builtin_nontemporal_load(src + 7 * 4);

    __builtin_nontemporal_store(r0, dst + 0 * W4);
    __builtin_nontemporal_store(r1, dst + 1 * W4);
    __builtin_nontemporal_store(r2, dst + 2 * W4);
    __builtin_nontemporal_store(r3, dst + 3 * W4);
    __builtin_nontemporal_store(r4, dst + 4 * W4);
    __builtin_nontemporal_store(r5, dst + 5 * W4);
    __builtin_nontemporal_store(r6, dst + 6 * W4);
    __builtin_nontemporal_store(r7, dst + 7 * W4);
}

extern "C" void kernel_launch(void* const* d_in, const int* in_sizes, int n_in,
                              void* d_out, int out_size, void* d_ws, size_t ws_size,
                              hipStream_t stream)
{
    (void)in_sizes; (void)n_in; (void)out_size; (void)d_ws; (void)ws_size;

    const float* patches = (const float*)d_in[0];  // [8,3,4096,16,16] f32
    float*       out     = (float*)d_out;          // [8,3,1016,1016] f32

    // grid.z = B*C = 24, grid.y = HOUT/8 = 127 (8 rows/thread),
    // block.x = 256 lanes covering 254 float4 columns.
    dim3 grid(1, 127, 24);
    dim3 block(256, 1, 1);
    recons_tensor_kernel<<<grid, block, 0, stream>>>(patches, out);
}